// YoloPredictLayer_43731357007999
// MI455X (gfx1250) — compile-verified
//
#include <hip/hip_runtime.h>
#include <math.h>

#define BATCH    32
#define KDET     512
#define NCLS     80
#define NCH      85
#define CAND_MAX 1024
#define NTHREADS 512
#define SCAN_THREADS 256
#define SCAN_SLICES  32
#define NMS_THR  0.5f
// sigmoid(x) >= 0.9  <=>  x >= log(0.9/0.1)
#define LOGIT_THR 2.1972246f

typedef float v2f __attribute__((ext_vector_type(2)));
typedef float v8f __attribute__((ext_vector_type(8)));

__device__ __forceinline__ float sigmoidf_(float x) {
    return 1.0f / (1.0f + __expf(-x));
}

// ---- Kernel 0: zero the per-batch candidate counters in workspace ----
__global__ void zero_counters_kernel(int* __restrict__ cnt) {
    if (threadIdx.x < BATCH) cnt[threadIdx.x] = 0;
}

// ---- Kernel A: wide scan of the objectness channel; append candidates ----
// grid = (SCAN_SLICES, BATCH); spreads strided cacheline traffic over many WGPs.
__global__ __launch_bounds__(SCAN_THREADS)
void scan_kernel(const float* __restrict__ pred, int N,
                 int* __restrict__ gcnt, float* __restrict__ gscore, int* __restrict__ gidx)
{
    const int b = blockIdx.y;
    const float* pred_b = pred + (size_t)b * (size_t)N * NCH;
    const int sliceLen = (N + SCAN_SLICES - 1) / SCAN_SLICES;
    const int a0 = blockIdx.x * sliceLen;
    const int a1 = min(a0 + sliceLen, N);

    for (int a = a0 + threadIdx.x; a < a1; a += SCAN_THREADS) {
        if (a + 2 * SCAN_THREADS < a1)
            __builtin_prefetch(&pred_b[(size_t)(a + 2 * SCAN_THREADS) * NCH + 4], 0, 1);
        float x = pred_b[(size_t)a * NCH + 4];       // objectness logit
        if (x >= LOGIT_THR) {                        // == sigmoid(x) >= 0.9
            int p = atomicAdd(&gcnt[b], 1);
            if (p < CAND_MAX) {
                gscore[b * CAND_MAX + p] = x;
                gidx[b * CAND_MAX + p]   = a;
            }
        }
    }
}

// ---- Kernel B: per-batch sort + gather + WMMA IoU suppress matrix + NMS ----
// If gcnt==nullptr, performs its own scan (fallback when ws is too small).
__global__ __launch_bounds__(NTHREADS)
void yolo_nms_kernel(const float* __restrict__ pred, float* __restrict__ out, int N,
                     const int* __restrict__ gcnt, const float* __restrict__ gscore,
                     const int* __restrict__ gidx)
{
    __shared__ float cs[CAND_MAX];      // candidate objectness LOGITS (order == sigmoid order)
    __shared__ int   ci[CAND_MAX];
    __shared__ float bx1[KDET], by1[KDET], bx2[KDET], by2[KDET];
    __shared__ float barea[KDET], bobj[KDET], bconf[KDET];
    __shared__ int   bcls[KDET];
    __shared__ unsigned supw[KDET * 16];   // 512 rows x 512-bit suppress mask
    __shared__ unsigned keepw[16];         // 512-bit keep mask
    __shared__ int nc;

    const int tid = threadIdx.x;
    const int b   = blockIdx.x;
    const float* pred_b = pred + (size_t)b * (size_t)N * NCH;

    if (tid == 0) nc = 0;
    if (tid < 16) keepw[tid] = 0u;
    __syncthreads();

    int count;
    if (gcnt) {
        // ---- Phase 1a: load candidate list produced by scan_kernel ----
        count = gcnt[b];
        count = count < CAND_MAX ? count : CAND_MAX;
        for (int p = tid; p < CAND_MAX; p += NTHREADS) {
            if (p < count) { cs[p] = gscore[b * CAND_MAX + p]; ci[p] = gidx[b * CAND_MAX + p]; }
            else           { cs[p] = -3.0e38f;                 ci[p] = 0x7FFFFFFF; }
        }
        __syncthreads();
    } else {
        // ---- Phase 1b (fallback): self-scan ----
        for (int a = tid; a < N; a += NTHREADS) {
            if (a + 2 * NTHREADS < N)
                __builtin_prefetch(&pred_b[(size_t)(a + 2 * NTHREADS) * NCH + 4], 0, 1);
            float x = pred_b[(size_t)a * NCH + 4];
            if (x >= LOGIT_THR) {
                int p = atomicAdd(&nc, 1);
                if (p < CAND_MAX) { cs[p] = x; ci[p] = a; }
            }
        }
        __syncthreads();
        count = nc < CAND_MAX ? nc : CAND_MAX;
        for (int p = count + tid; p < CAND_MAX; p += NTHREADS) { cs[p] = -3.0e38f; ci[p] = 0x7FFFFFFF; }
    }

    // ---- Phase 2: bitonic sort 1024 slots by (logit desc, index asc) == top_k order ----
    for (unsigned kk = 2; kk <= CAND_MAX; kk <<= 1) {
        for (unsigned j = kk >> 1; j > 0; j >>= 1) {
            __syncthreads();
            for (unsigned i = tid; i < CAND_MAX; i += NTHREADS) {
                unsigned ixj = i ^ j;
                if (ixj > i) {
                    float si = cs[i], sj = cs[ixj];
                    int   ii = ci[i], ij = ci[ixj];
                    bool iBefore    = (si > sj) || ((si == sj) && (ii < ij));
                    bool wantIFirst = ((i & kk) == 0);
                    if (iBefore != wantIFirst) {
                        cs[i] = sj; cs[ixj] = si; ci[i] = ij; ci[ixj] = ii;
                    }
                }
            }
        }
    }
    __syncthreads();

    // ---- Phase 3: gather full 85-float rows only for survivors ----
    {
        int k = tid;                               // NTHREADS == KDET
        bool valid = (k < count) && (cs[k] >= LOGIT_THR);
        float x1 = 0, y1 = 0, x2 = 0, y2 = 0, objv = 0, confv = 0;
        int cid = 0;
        if (valid) {
            const float* p = pred_b + (size_t)ci[k] * NCH;
            float cx = p[0], cy = p[1], w = p[2], h = p[3];
            x1 = cx - w * 0.5f; y1 = cy - h * 0.5f;
            x2 = cx + w * 0.5f; y2 = cy + h * 0.5f;
            objv = sigmoidf_(p[4]);
            float best = -1.0f; int bi = 0;
            #pragma unroll 4
            for (int c = 0; c < NCLS; ++c) {
                float sc = sigmoidf_(p[5 + c]);
                if (sc > best) { best = sc; bi = c; }
            }
            confv = best; cid = bi;
        }
        bx1[k] = x1; by1[k] = y1; bx2[k] = x2; by2[k] = y2;
        barea[k] = fmaxf(x2 - x1, 0.f) * fmaxf(y2 - y1, 0.f);
        bobj[k] = objv; bconf[k] = confv; bcls[k] = cid;
        if (valid) atomicOr(&keepw[k >> 5], 1u << (k & 31));
        #pragma unroll
        for (int w2 = 0; w2 < 16; ++w2) supw[k * 16 + w2] = 0u;
    }
    __syncthreads();

    // ---- Phase 4: suppress matrix; area[i]+area[j] via V_WMMA_F32_16X16X4_F32,
    //      division-free IoU test: iou > t  <=>  inter > t * union ----
    {
        const int wave = tid >> 5;
        const int lane = tid & 31;
        for (int t = wave; t < 32 * 32; t += NTHREADS / 32) {
            int ti = t >> 5, tj = t & 31;
            int i0 = ti * 16, j0 = tj * 16;
            v2f a, bb;
            a.x  = (lane < 16) ? barea[i0 + lane] : 0.0f;   // A row m = [area_i, 1, 0, 0]
            a.y  = (lane < 16) ? 1.0f              : 0.0f;
            bb.x = (lane < 16) ? 1.0f              : 0.0f;  // B col n = [1, area_j, 0, 0]^T
            bb.y = (lane < 16) ? barea[j0 + lane] : 0.0f;
            v8f cz = {0.f, 0.f, 0.f, 0.f, 0.f, 0.f, 0.f, 0.f};
            v8f d = __builtin_amdgcn_wmma_f32_16x16x4_f32(
                        false, a, false, bb, (short)0, cz, false, false);
            int rbase = i0 + ((lane >= 16) ? 8 : 0);
            int col   = j0 + (lane & 15);
            float cx1 = bx1[col], cy1 = by1[col], cx2 = bx2[col], cy2 = by2[col];
            int ccls = bcls[col];
            #pragma unroll
            for (int v = 0; v < 8; ++v) {
                int row = rbase + v;
                float xx1 = fmaxf(bx1[row], cx1);
                float yy1 = fmaxf(by1[row], cy1);
                float xx2 = fminf(bx2[row], cx2);
                float yy2 = fminf(by2[row], cy2);
                float inter = fmaxf(xx2 - xx1, 0.f) * fmaxf(yy2 - yy1, 0.f);
                float uni   = fmaxf(d[v] - inter, 1e-9f);   // d[v] = area[row]+area[col]
                // iou > NMS_THR  <=>  inter > NMS_THR * uni   (uni > 0) -- no division
                if ((inter > NMS_THR * uni) && (bcls[row] == ccls) && (row != col))
                    atomicOr(&supw[row * 16 + (col >> 5)], 1u << (col & 31));
            }
        }
    }
    __syncthreads();

    // ---- Phase 5: sequential greedy NMS on bitmasks, single wave32 ----
    if (tid < 32) {
        volatile unsigned* vkeep = keepw;
        const int lane = tid;
        for (int i = 0; i < KDET; ++i) {
            unsigned kw = vkeep[i >> 5];               // uniform read across wave
            if (kw & (1u << (i & 31))) {
                if (lane < 16) {
                    unsigned iw  = (unsigned)(i >> 5), ib2 = (unsigned)(i & 31);
                    unsigned later = ((unsigned)lane > iw) ? 0xFFFFFFFFu
                                   : (((unsigned)lane == iw) ? (0xFFFFFFFEu << ib2) : 0u);
                    unsigned s = supw[i * 16 + lane] & later;
                    if (s) vkeep[lane] = vkeep[lane] & ~s;
                }
            }
            __builtin_amdgcn_wave_barrier();           // same-wave DS ops are in-order
        }
    }
    __syncthreads();

    // ---- Phase 6: emit dets * keep and keep mask ----
    {
        int k = tid;
        unsigned kb = (keepw[k >> 5] >> (k & 31)) & 1u;
        float m = (float)kb;
        float* dets  = out;
        float* keepf = out + (size_t)BATCH * KDET * 7;
        size_t ob = ((size_t)b * KDET + k) * 7;
        dets[ob + 0] = bx1[k] * m;
        dets[ob + 1] = by1[k] * m;
        dets[ob + 2] = bx2[k] * m;
        dets[ob + 3] = by2[k] * m;
        dets[ob + 4] = bobj[k] * m;
        dets[ob + 5] = bconf[k] * m;
        dets[ob + 6] = (float)bcls[k] * m;
        keepf[(size_t)b * KDET + k] = m;
    }
}

extern "C" void kernel_launch(void* const* d_in, const int* in_sizes, int n_in,
                              void* d_out, int out_size, void* d_ws, size_t ws_size,
                              hipStream_t stream) {
    (void)n_in; (void)out_size;
    const float* pred = (const float*)d_in[0];
    // d_in[1] is max_det (=512); KDET is compile-time (sizes LDS), matching the reference.
    int N = in_sizes[0] / (BATCH * NCH);   // 22743

    // Workspace layout: [int cnt[32]] [float score[32][1024]] [int idx[32][1024]]
    const size_t needWs = BATCH * sizeof(int)
                        + (size_t)BATCH * CAND_MAX * sizeof(float)
                        + (size_t)BATCH * CAND_MAX * sizeof(int);

    if (d_ws && ws_size >= needWs) {
        int*   cnt    = (int*)d_ws;
        float* wscore = (float*)(cnt + BATCH);
        int*   widx   = (int*)(wscore + (size_t)BATCH * CAND_MAX);
        zero_counters_kernel<<<1, 32, 0, stream>>>(cnt);
        dim3 sgrid(SCAN_SLICES, BATCH);
        scan_kernel<<<sgrid, SCAN_THREADS, 0, stream>>>(pred, N, cnt, wscore, widx);
        yolo_nms_kernel<<<BATCH, NTHREADS, 0, stream>>>(pred, (float*)d_out, N,
                                                        cnt, wscore, widx);
    } else {
        yolo_nms_kernel<<<BATCH, NTHREADS, 0, stream>>>(pred, (float*)d_out, N,
                                                        nullptr, nullptr, nullptr);
    }
}